// AggregateLayer_64725157151122
// MI455X (gfx1250) — compile-verified
//
#include <hip/hip_runtime.h>
#include <hip/hip_bf16.h>
#include <math.h>

typedef __attribute__((ext_vector_type(2))) float v2f;
typedef __attribute__((ext_vector_type(8))) float v8f;

#define ORD_NEG_INF 0x007FFFFFu  // f2ord(-inf)

__device__ __forceinline__ unsigned f2ord(float f) {
  unsigned u = __float_as_uint(f);
  return (u & 0x80000000u) ? ~u : (u | 0x80000000u);
}
__device__ __forceinline__ float ord2f(unsigned u) {
  return (u & 0x80000000u) ? __uint_as_float(u & 0x7FFFFFFFu)
                           : __uint_as_float(~u);
}

__device__ __forceinline__ float v8_get(v8f v, int i) {
  float r = v[0];
  r = (i == 1) ? v[1] : r;
  r = (i == 2) ? v[2] : r;
  r = (i == 3) ? v[3] : r;
  r = (i == 4) ? v[4] : r;
  r = (i == 5) ? v[5] : r;
  r = (i == 6) ? v[6] : r;
  r = (i == 7) ? v[7] : r;
  return r;
}

// ---------- Stage 1 ----------

__global__ void k_init(float* __restrict__ H, unsigned* __restrict__ Mo,
                       float* __restrict__ S, long nH, int nRN) {
  long i = (long)blockIdx.x * blockDim.x + threadIdx.x;
  if (i < nH) H[i] = 0.0f;
  if (i < nRN) { Mo[i] = ORD_NEG_INF; S[i] = 0.0f; }
}

__global__ void k_logit_max(const float* __restrict__ d,
                            const float* __restrict__ d1,
                            const float* __restrict__ d2,
                            const int* __restrict__ src,
                            const int* __restrict__ dst,
                            const int* __restrict__ split_p,
                            float* __restrict__ L, unsigned* __restrict__ Mo,
                            int E, int Nsrc, int Nn, int R) {
  int i = blockIdx.x * blockDim.x + threadIdx.x;
  if (i >= R * E) return;
  int r = i / E;
  int si = src[i];
  int dn = dst[i];
  int split = *split_p;
  float dd = d[r * Nsrc + si];
  float l = (dn < split) ? (d1[r * Nsrc + si] / dd) : (-d2[r * Nsrc + si] / dd);
  L[i] = l;
  atomicMax(&Mo[r * Nn + dn], f2ord(l));
}

__global__ void k_fin_m(unsigned* __restrict__ Mo, int n) {
  int i = blockIdx.x * blockDim.x + threadIdx.x;
  if (i >= n) return;
  float m = ord2f(Mo[i]);
  if (m < -1.0e30f) m = 0.0f;  // nodes with no in-edges: max = -inf -> 0
  ((float*)Mo)[i] = m;
}

__global__ void k_exp_sum(float* __restrict__ L, const float* __restrict__ Mf,
                          float* __restrict__ S, const int* __restrict__ dst,
                          int E, int Nn, int R) {
  int i = blockIdx.x * blockDim.x + threadIdx.x;
  if (i >= R * E) return;
  int r = i / E;
  int dn = dst[i];
  float ev = expf(L[i] - Mf[r * Nn + dn]);
  L[i] = ev;
  atomicAdd(&S[r * Nn + dn], ev);
}

// one wave (32 lanes) per edge; D=256 -> 2 float4 per lane
__global__ void k_scatter(const float* __restrict__ L,
                          const float* __restrict__ S,
                          const float* __restrict__ x,
                          const int* __restrict__ src,
                          const int* __restrict__ dst,
                          float* __restrict__ H,
                          int E, int Nsrc, int Nn, int R, int D) {
  int lane = threadIdx.x & 31;
  int widx = blockIdx.x * (blockDim.x >> 5) + (threadIdx.x >> 5);
  if (widx >= R * E) return;
  int r = widx / E;
  int si = src[widx];
  int dn = dst[widx];
  float a = L[widx] / fmaxf(S[r * Nn + dn], 1e-9f);
  const float4* xr = (const float4*)(x + (long)(r * Nsrc + si) * D);
  float* hr = H + ((long)dn * R + r) * D;
  for (int j = lane; j < (D >> 2); j += 32) {
    float4 v = xr[j];
    atomicAdd(&hr[4 * j + 0], a * v.x);
    atomicAdd(&hr[4 * j + 1], a * v.y);
    atomicAdd(&hr[4 * j + 2], a * v.z);
    atomicAdd(&hr[4 * j + 3], a * v.w);
  }
}

// ---------- Stage 2: candidate mask via WMMA diff-Gram diagonal ----------
// Block = 256 threads = 8 waves, one block per node. RD = R*D = 1024.
// Pass 1 computes ssq[k] on the matrix unit AND stages all diffs in LDS so
// pass 2 never touches global memory for candidate rows.
__global__ void __launch_bounds__(256)
k_mask_out(const float* __restrict__ H, const int* __restrict__ cand,
           float* __restrict__ out, int Nn) {
  constexpr int RD = 1024;  // R=4, D=256
  constexpr int Kc = 16;
  constexpr int CT = RD + 4;  // padded stride: 4*ck bank offset -> conflict-free
  __shared__ float ctile[Kc * CT];  // diff values, 65.8 KB
  __shared__ float hn[RD];
  __shared__ float red[8 * Kc];
  __shared__ float ssq_s[Kc];
  __shared__ int cnd[Kc];

  int n = blockIdx.x;
  int t = threadIdx.x;
  for (int i = t; i < RD; i += 256) hn[i] = H[(long)n * RD + i];
  if (t < Kc) cnd[t] = cand[n * Kc + t];
  __syncthreads();

  int w = t >> 5;
  int lane = t & 31;
  int ck = lane & 15;            // candidate this lane feeds into A/B
  int koff = (lane >> 4) * 2;    // K sub-offset: lanes 0-15 -> {0,1}, 16-31 -> {2,3}
  const float* crow = H + (long)cnd[ck] * RD;
  int cbase = ck * CT;
  // pull this wave's slice of the candidate row toward cache
  __builtin_prefetch(crow + w * 128, 0, 0);

  // Pass 1: C += diff_tile * diff_tile^T  (16x16x4 f32 WMMA); diag(C) = ssq
  v8f acc = {};
  int c0 = w * 32;
  for (int c = c0; c < c0 + 32; ++c) {
    int base = c * 4 + koff;
    v2f xv;
    xv.x = crow[base] - hn[base];
    xv.y = crow[base + 1] - hn[base + 1];
    *(v2f*)&ctile[cbase + base] = xv;  // stage diff for pass 2 (8B aligned)
    acc = __builtin_amdgcn_wmma_f32_16x16x4_f32(
        /*neg_a=*/false, xv, /*neg_b=*/false, xv,
        /*c_mod=*/(short)0, acc, /*reuse_a=*/false, /*reuse_b=*/false);
  }
  // diagonal (m,m): m<8 -> VGPR m, lane m ; m>=8 -> VGPR m-8, lane m+16
  if (lane < 8) {
    red[w * Kc + lane] = v8_get(acc, lane);
  } else if (lane >= 24) {
    red[w * Kc + (lane - 16)] = v8_get(acc, lane - 24);
  }
  __syncthreads();
  if (t < Kc) {
    float s = 0.0f;
    for (int ww = 0; ww < 8; ++ww) s += red[ww * Kc + t];
    ssq_s[t] = s;
  }
  __syncthreads();

  // softmax over K (each thread redundantly; 16 values)
  float att[Kc];
  float mx = -INFINITY;
  for (int k = 0; k < Kc; ++k) {
    att[k] = -sqrtf(ssq_s[k]);
    mx = fmaxf(mx, att[k]);
  }
  float se = 0.0f;
  for (int k = 0; k < Kc; ++k) { att[k] = expf(att[k] - mx); se += att[k]; }
  float inv = 1.0f / se;
  for (int k = 0; k < Kc; ++k) att[k] *= inv;

  // Pass 2: thread t owns output dim d = t; rd = t + 256*r; diffs from LDS
  float o = 0.0f;
  for (int r = 0; r < 4; ++r) {
    int rd = t + (r << 8);
    float hv = hn[rd];
    float wsum = 0.0f;
    for (int k = 0; k < Kc; ++k) {
      float diff = ctile[k * CT + rd];
      wsum += att[k] * diff * diff;
    }
    o += hv * expf(-wsum);
  }
  out[(long)n * 256 + t] = o;
}

// ---------- Host launch ----------

extern "C" void kernel_launch(void* const* d_in, const int* in_sizes, int n_in,
                              void* d_out, int out_size, void* d_ws, size_t ws_size,
                              hipStream_t stream) {
  const float* x = (const float*)d_in[0];
  const float* d = (const float*)d_in[1];
  const float* d1 = (const float*)d_in[2];
  const float* d2 = (const float*)d_in[3];
  const int* src = (const int*)d_in[4];
  const int* dst = (const int*)d_in[5];
  const int* cand = (const int*)d_in[6];
  const int* split = (const int*)d_in[7];

  const int R = 4, Kc = 16, D = 256;
  const int Nn = in_sizes[6] / Kc;      // 20000
  const int E = in_sizes[4] / R;        // 640000
  const int Nsrc = in_sizes[1] / R;     // 20000

  float* ws = (float*)d_ws;
  long nH = (long)Nn * R * D;           // 20.48M floats
  long nRE = (long)R * E;               // 2.56M
  float* H = ws;
  float* L = H + nH;
  float* M = L + nRE;                   // ordered-uint max, then float m
  float* S = M + (long)R * Nn;

  int blk = 256;
  long gInit = (nH + blk - 1) / blk;
  k_init<<<(unsigned)gInit, blk, 0, stream>>>(H, (unsigned*)M, S, nH, R * Nn);

  long gRE = (nRE + blk - 1) / blk;
  k_logit_max<<<(unsigned)gRE, blk, 0, stream>>>(d, d1, d2, src, dst, split, L,
                                                 (unsigned*)M, E, Nsrc, Nn, R);
  k_fin_m<<<(R * Nn + blk - 1) / blk, blk, 0, stream>>>((unsigned*)M, R * Nn);
  k_exp_sum<<<(unsigned)gRE, blk, 0, stream>>>(L, M, S, dst, E, Nn, R);

  long gScat = (nRE + 7) / 8;  // 8 waves per 256-thread block, 1 wave/edge
  k_scatter<<<(unsigned)gScat, blk, 0, stream>>>(L, S, x, src, dst, H, E, Nsrc,
                                                 Nn, R, D);

  k_mask_out<<<Nn, 256, 0, stream>>>(H, cand, (float*)d_out, Nn);
}